// MyGraphSAGE_26663156973655
// MI455X (gfx1250) — compile-verified
//
#include <hip/hip_runtime.h>

#define HID 128
#define OUTD 64

typedef float v2f __attribute__((ext_vector_type(2)));
typedef float v8f __attribute__((ext_vector_type(8)));

// ---------------- degree kernels ----------------
__global__ __launch_bounds__(256) void k_deg(const int* __restrict__ dst,
                                             float* __restrict__ deg, int E) {
  int e = blockIdx.x * 256 + threadIdx.x;
  if (e < E) atomicAdd(&deg[dst[e]], 1.0f);
}

__global__ __launch_bounds__(256) void k_rdeg(float* __restrict__ deg, int N) {
  int i = blockIdx.x * 256 + threadIdx.x;
  if (i < N) deg[i] = 1.0f / fmaxf(deg[i], 1.0f);
}

// ---------------- edge-parallel gather / atomic scatter-sum ----------------
// D/4 consecutive threads handle one edge's D floats (float4 chunks) -> coalesced
template<int D>
__global__ __launch_bounds__(256) void k_agg(const float* __restrict__ feat,
                                             const int* __restrict__ src,
                                             const int* __restrict__ dst,
                                             float* __restrict__ sum, int E) {
  const int CH = D / 4;
  long long t = (long long)blockIdx.x * 256 + threadIdx.x;
  int e = (int)(t / CH);
  int c = (int)(t % CH);
  if (e >= E) return;
  int s = src[e], d = dst[e];
  const float4 v = *(const float4*)(feat + (size_t)s * D + (size_t)c * 4);
  float* o = sum + (size_t)d * D + (size_t)c * 4;
  atomicAdd(o + 0, v.x);
  atomicAdd(o + 1, v.y);
  atomicAdd(o + 2, v.z);
  atomicAdd(o + 3, v.w);
}

// ---------------- fused WMMA GEMM ----------------
// out[r,c] = act( A1@B1 (+ (A2 .* rdeg_row)@B2) + bias + addmat .* rdeg_row )
// K = HID = 128 for all layers; one wave computes one 16x16 tile.
template<int NCOLS, bool TWO, bool ADD_MEAN, bool RELU, bool BIAS>
__global__ __launch_bounds__(256) void k_gemm(const float* __restrict__ A1,
                                              const float* __restrict__ A2,
                                              const float* __restrict__ rdeg,
                                              const float* __restrict__ B1,
                                              const float* __restrict__ B2,
                                              const float* __restrict__ bias,
                                              const float* __restrict__ addmat,
                                              float* __restrict__ out, int M) {
  const int NT = NCOLS / 16;
  int wave = (blockIdx.x * 256 + (int)threadIdx.x) >> 5;
  int mt = wave / NT;
  int nt = wave - mt * NT;
  int row0 = mt * 16;
  if (row0 >= M) return;  // wave-uniform: EXEC stays all-1s for WMMA

  int lane = threadIdx.x & 31;
  int hh = lane >> 4;     // lane half selects K sub-pair / row upper half
  int lr = lane & 15;
  int col0 = nt * 16;

  // A fragment: lane holds row (row0+lr), Ks {k+2*hh, k+2*hh+1}
  const float* a1 = A1 + (size_t)(row0 + lr) * HID + 2 * hh;
  // B fragment: lane holds col (col0+lr), rows {k+2*hh, k+2*hh+1}
  const float* b1 = B1 + (size_t)(2 * hh) * NCOLS + col0 + lr;
  const float* a2 = nullptr;
  const float* b2 = nullptr;
  float rd = 0.f;
  if (TWO) {
    a2 = A2 + (size_t)(row0 + lr) * HID + 2 * hh;
    b2 = B2 + (size_t)(2 * hh) * NCOLS + col0 + lr;
    rd = rdeg[row0 + lr];  // fold mean (1/deg) into A2 fragment
  }

  v8f c = {};
#pragma unroll 4
  for (int k = 0; k < HID; k += 4) {
    v2f av = *(const v2f*)(a1 + k);
    v2f bv;
    bv.x = b1[(size_t)k * NCOLS];
    bv.y = b1[(size_t)(k + 1) * NCOLS];
    c = __builtin_amdgcn_wmma_f32_16x16x4_f32(false, av, false, bv, (short)0, c,
                                              false, false);
    if (TWO) {
      v2f av2 = *(const v2f*)(a2 + k);
      av2.x *= rd;
      av2.y *= rd;
      v2f bv2;
      bv2.x = b2[(size_t)k * NCOLS];
      bv2.y = b2[(size_t)(k + 1) * NCOLS];
      c = __builtin_amdgcn_wmma_f32_16x16x4_f32(false, av2, false, bv2, (short)0,
                                                c, false, false);
    }
  }

  float bb = BIAS ? bias[col0 + lr] : 0.f;
#pragma unroll
  for (int v = 0; v < 8; ++v) {
    int r = row0 + v + 8 * hh;  // C/D layout: VGPR v holds rows v and v+8
    float val = c[v] + bb;
    if (ADD_MEAN) val += addmat[(size_t)r * NCOLS + col0 + lr] * rdeg[r];
    if (RELU) val = fmaxf(val, 0.f);
    out[(size_t)r * NCOLS + col0 + lr] = val;
  }
}

// ---------------- launcher ----------------
extern "C" void kernel_launch(void* const* d_in, const int* in_sizes, int n_in,
                              void* d_out, int out_size, void* d_ws,
                              size_t ws_size, hipStream_t stream) {
  const float* x   = (const float*)d_in[0];
  const int*   src = (const int*)d_in[1];
  const int*   dst = (const int*)d_in[2];
  const float* Ws1 = (const float*)d_in[3];
  const float* b1  = (const float*)d_in[4];
  const float* Wn1 = (const float*)d_in[5];
  const float* Ws2 = (const float*)d_in[6];
  const float* b2  = (const float*)d_in[7];
  const float* Wn2 = (const float*)d_in[8];
  const float* Ws3 = (const float*)d_in[9];
  const float* b3  = (const float*)d_in[10];
  const float* Wn3 = (const float*)d_in[11];
  float* out = (float*)d_out;

  const int N = in_sizes[0] / HID;
  const int E = in_sizes[1];

  // workspace layout (floats): deg[N] | h1[N*128] | h2[N*128] | t[N*64] | sum[N*128]
  float* deg = (float*)d_ws;
  float* h1  = deg + N;
  float* h2  = h1 + (size_t)N * HID;
  float* tb  = h2 + (size_t)N * HID;
  float* sum = tb + (size_t)N * OUTD;

  const int mtiles = (N + 15) / 16;
  const int tilesH = mtiles * (HID / 16);
  const int tilesO = mtiles * (OUTD / 16);
  const long long aggTH = (long long)E * (HID / 4);
  const long long aggTO = (long long)E * (OUTD / 4);

  // degrees (shared by all layers)
  hipMemsetAsync(deg, 0, (size_t)N * sizeof(float), stream);
  hipMemsetAsync(sum, 0, (size_t)N * HID * sizeof(float), stream);
  k_deg<<<(E + 255) / 256, 256, 0, stream>>>(dst, deg, E);
  k_rdeg<<<(N + 255) / 256, 256, 0, stream>>>(deg, N);

  // layer 1: h1 = relu(x@Ws1 + b1 + mean(x)@Wn1)
  k_agg<HID><<<(int)((aggTH + 255) / 256), 256, 0, stream>>>(x, src, dst, sum, E);
  k_gemm<HID, true, false, true, true><<<(tilesH + 7) / 8, 256, 0, stream>>>(
      x, sum, deg, Ws1, Wn1, b1, nullptr, h1, N);

  // layer 2: h2 = relu(h1@Ws2 + b2 + mean(h1)@Wn2)
  hipMemsetAsync(sum, 0, (size_t)N * HID * sizeof(float), stream);
  k_agg<HID><<<(int)((aggTH + 255) / 256), 256, 0, stream>>>(h1, src, dst, sum, E);
  k_gemm<HID, true, false, true, true><<<(tilesH + 7) / 8, 256, 0, stream>>>(
      h1, sum, deg, Ws2, Wn2, b2, nullptr, h2, N);

  // layer 3 (lin_before_mp): t = h2@Wn3 ; out = h2@Ws3 + b3 + mean(t)
  k_gemm<OUTD, false, false, false, false><<<(tilesO + 7) / 8, 256, 0, stream>>>(
      h2, nullptr, nullptr, Wn3, nullptr, nullptr, nullptr, tb, N);
  hipMemsetAsync(sum, 0, (size_t)N * OUTD * sizeof(float), stream);
  k_agg<OUTD><<<(int)((aggTO + 255) / 256), 256, 0, stream>>>(tb, src, dst, sum, E);
  k_gemm<OUTD, false, true, false, true><<<(tilesO + 7) / 8, 256, 0, stream>>>(
      h2, nullptr, deg, Ws3, nullptr, b3, sum, out, N);
}